// LocallyConnectedXYZLayer_9320079032367
// MI455X (gfx1250) — compile-verified
//
#include <hip/hip_runtime.h>

// Problem constants (from reference)
#define NB 2          // batch
#define NC 20         // classes/channels
#define NH 64         // height
#define NW 2048       // width (power of two -> wrap via & (NW-1))
#define RR 5          // window rows
#define TW 128        // W-tile per workgroup == blockDim.x
#define COLS (TW + 4) // tile columns incl. +-2 halo

// ---------------------------------------------------------------------------
// CDNA5 async global->LDS path (ASYNCcnt-tracked). Guarded so the compile log
// tells us which path was taken on this toolchain.
// ---------------------------------------------------------------------------
#if defined(__HIP_DEVICE_COMPILE__) && __has_builtin(__builtin_amdgcn_global_load_async_to_lds_b32)
#define HAVE_ASYNC_LDS 1
#warning "CDNA5 path: using __builtin_amdgcn_global_load_async_to_lds_b32"
#else
#define HAVE_ASYNC_LDS 0
#if defined(__HIP_DEVICE_COMPILE__)
#warning "CDNA5 path: async-to-LDS builtin NOT available, using sync fallback"
#endif
#endif

#if HAVE_ASYNC_LDS
typedef __attribute__((address_space(1))) int g_as_int;
typedef __attribute__((address_space(3))) int l_as_int;
#endif

__device__ __forceinline__ void g2l_b32(const float* g, float* l) {
#if HAVE_ASYNC_LDS
  __builtin_amdgcn_global_load_async_to_lds_b32(
      (g_as_int*)g, (l_as_int*)l, /*imm offset=*/0, /*cpol=*/0);
#else
  *l = __builtin_nontemporal_load(g); // plain fallback: load + ds_store
#endif
}

__device__ __forceinline__ void wait_async_lds() {
#if HAVE_ASYNC_LDS
#if __has_builtin(__builtin_amdgcn_s_wait_asynccnt)
  __builtin_amdgcn_s_wait_asynccnt(0);
#else
  asm volatile("s_wait_asynccnt 0x0" ::: "memory");
#endif
#endif
}

// ---------------------------------------------------------------------------
// Kernel: one workgroup = one (n, h, 128-wide W slice); one thread = one pixel.
// ---------------------------------------------------------------------------
__global__ __launch_bounds__(TW) void lcxyz_kernel(
    const float* __restrict__ xyz,      // (NB, 3, NH, NW)
    const float* __restrict__ softmax,  // (NB, NC, NH, NW)
    const int*   __restrict__ mask,     // (NB, NH, NW)
    float*       __restrict__ out)      // (NB, NC, NH, NW)
{
  // LDS staging (total 66000 B, ~64.5 KB):
  __shared__ float  s_sm [RR * COLS * NC]; // raw softmax, channel-contiguous (80B/site)
  __shared__ float4 s_xyz[RR * COLS];      // x,y,z interleaved (w unused)
  __shared__ float  s_msk[RR * COLS];      // mask as float (0 for invalid rows)

  const int tid = threadIdx.x;
  const int n   = blockIdx.z;
  const int h   = blockIdx.y;
  const int w0  = blockIdx.x * TW;

  // ---- stage xyz + mask (3 async b32 per site; zero-fill invalid rows) ----
  for (int idx = tid; idx < RR * COLS; idx += TW) {
    const int r   = idx / COLS;
    const int col = idx - r * COLS;
    const int hh  = h - 2 + r;
    const int ww  = (w0 + col - 2 + NW) & (NW - 1); // column wrap
    float* dst = (float*)&s_xyz[idx];
    if ((unsigned)hh < (unsigned)NH) {
      const size_t gb = ((size_t)n * 3 * NH + hh) * (size_t)NW + ww; // x-plane
      g2l_b32(xyz + gb,                          dst + 0);
      g2l_b32(xyz + gb + (size_t)NH * NW,        dst + 1);
      g2l_b32(xyz + gb + (size_t)2 * NH * NW,    dst + 2);
      s_msk[idx] = (float)mask[((size_t)n * NH + hh) * (size_t)NW + ww];
    } else {
      dst[0] = 0.0f; dst[1] = 0.0f; dst[2] = 0.0f;
      s_msk[idx] = 0.0f;
    }
  }

  // ---- stage raw softmax tile: [r][col][c], c contiguous ----
  for (int idx = tid; idx < RR * COLS * NC; idx += TW) {
    const int rc  = idx / NC;          // site index r*COLS+col
    const int c   = idx - rc * NC;
    const int r   = rc / COLS;
    const int col = rc - r * COLS;
    const int hh  = h - 2 + r;
    if ((unsigned)hh < (unsigned)NH) {
      const int ww = (w0 + col - 2 + NW) & (NW - 1);
      const float* g = softmax + (((size_t)(n * NC + c) * NH + hh) * (size_t)NW + ww);
      g2l_b32(g, &s_sm[idx]);
    } else {
      s_sm[idx] = 0.0f; // invalid rows contribute exactly zero
    }
  }

  wait_async_lds();   // ASYNCcnt == 0 for this wave's DMAs
  __syncthreads();    // all waves' staging visible

  // ---- per-pixel compute ----
  const int lw = tid;                 // local pixel column
  const float4 cp = s_xyz[2 * COLS + lw + 2]; // center xyz (r=2, halo offset +2)

  float acc[NC];
#pragma unroll
  for (int c = 0; c < NC; ++c) acc[c] = 0.0f;

#pragma unroll
  for (int r = 0; r < RR; ++r) {
#pragma unroll
    for (int j = 0; j < 5; ++j) {
      const int site = r * COLS + lw + j;     // neighbor site in tile coords
      const float4 q = s_xyz[site];
      const float dx = q.x - cp.x;
      const float dy = q.y - cp.y;
      const float dz = q.z - cp.z;
      const float d2 = dx * dx + dy * dy + dz * dz;
      // gauss * mask folded once per site (mask==0 also kills invalid rows)
      const float g = __expf(-0.5f * d2) * s_msk[site];

      const float4* sp = (const float4*)&s_sm[site * NC]; // 80B-aligned -> b128
#pragma unroll
      for (int cg = 0; cg < NC / 4; ++cg) {
        const float4 v = sp[cg];
        acc[cg * 4 + 0] += g * v.x;
        acc[cg * 4 + 1] += g * v.y;
        acc[cg * 4 + 2] += g * v.z;
        acc[cg * 4 + 3] += g * v.w;
      }
    }
  }

  // ---- coalesced output stores ----
  const int w = w0 + lw;
#pragma unroll
  for (int c = 0; c < NC; ++c) {
    out[((size_t)(n * NC + c) * NH + h) * (size_t)NW + w] = acc[c];
  }
}

// ---------------------------------------------------------------------------
extern "C" void kernel_launch(void* const* d_in, const int* in_sizes, int n_in,
                              void* d_out, int out_size, void* d_ws, size_t ws_size,
                              hipStream_t stream) {
  (void)in_sizes; (void)n_in; (void)out_size; (void)d_ws; (void)ws_size;
  const float* xyz     = (const float*)d_in[0];
  const float* softmax = (const float*)d_in[1];
  const int*   mask    = (const int*)d_in[2];
  float*       out     = (float*)d_out;

  dim3 grid(NW / TW, NH, NB); // (16, 64, 2)
  dim3 block(TW);             // 128 threads = 4 wave32
  lcxyz_kernel<<<grid, block, 0, stream>>>(xyz, softmax, mask, out);
}